// euclid2d_49606872269313
// MI455X (gfx1250) — compile-verified
//
#include <hip/hip_runtime.h>

typedef __attribute__((ext_vector_type(2))) float v2f;
typedef __attribute__((ext_vector_type(4))) float v4f;
typedef __attribute__((ext_vector_type(8))) float v8f;

#define N_BATCH   8
#define C_IN      3
#define H_IMG     64
#define W_IMG     64
#define KSZ       5
#define PADV      2
#define NFILT     64
#define DDIM      75          // C_IN * KSZ * KSZ
#define DPAD      76          // padded to multiple of 4 (K-dim of WMMA f32 16x16x4)
#define M_TOTAL   (H_IMG * W_IMG * N_BATCH)   // 32768 columns
#define COLS_WG   32          // columns per workgroup
#define OUT_FSTR  (H_IMG * W_IMG)             // 4096
#define OUT_ISTR  (NFILT * H_IMG * W_IMG)     // 262144

__global__ __launch_bounds__(256)
void euclid2d_wmma_f32_kernel(const float* __restrict__ x,
                              const float* __restrict__ w,
                              float* __restrict__ out)
{
    // K-pair-interleaved staging: element (d, n) lives at pair[d/2][n].{x,y}[d&1].
    // This makes each lane's WMMA A/B fragment one aligned ds_load_b64, and the
    // per-kk address delta a compile-time DS offset immediate.
    __shared__ v2f XldsP[(DPAD / 2) * COLS_WG];   // 38*32*8B  = 9.5 KB
    __shared__ v2f WldsP[(DPAD / 2) * NFILT];     // 38*64*8B  = 19 KB
    __shared__ float Tile[NFILT * COLS_WG];       // 64*32*4B  = 8 KB (epilogue)
    __shared__ float x2s[COLS_WG];
    __shared__ float w2s[NFILT];

    float* Xf = (float*)XldsP;
    float* Wf = (float*)WldsP;

    const int tid   = threadIdx.x;
    const int mBase = blockIdx.x * COLS_WG;

    // ---- Cooperative load of X tile (unfold with pad=2 done here) ----
    for (int e = tid; e < DPAD * COLS_WG; e += 256) {
        const int d   = e >> 5;        // feature index
        const int col = e & 31;        // local column
        float v = 0.0f;
        if (d < DDIM) {
            const int m  = mBase + col;
            const int i  = m & 7;      // batch  (column index = p*n + i)
            const int p  = m >> 3;     // spatial position
            const int ho = p >> 6;
            const int wo = p & 63;
            const int c   = d / (KSZ * KSZ);
            const int rem = d - c * KSZ * KSZ;
            const int kh  = rem / KSZ;
            const int kw  = rem - kh * KSZ;
            const int r   = ho + kh - PADV;
            const int cc  = wo + kw - PADV;
            if (r >= 0 && r < H_IMG && cc >= 0 && cc < W_IMG)
                v = x[((i * C_IN + c) * H_IMG + r) * W_IMG + cc];
        }
        Xf[(d >> 1) * (2 * COLS_WG) + col * 2 + (d & 1)] = v;
    }

    // ---- Cooperative load of W, pair-interleaved [d/2][f]{d, d+1} ----
    for (int e = tid; e < DPAD * NFILT; e += 256) {
        const int d = e >> 6;
        const int f = e & 63;
        Wf[(d >> 1) * (2 * NFILT) + f * 2 + (d & 1)] = (d < DDIM) ? w[f * DDIM + d] : 0.0f;
    }
    __syncthreads();

    // ---- Norm terms: x2 per column (32 threads), w2 per filter (64 threads) ----
    if (tid < COLS_WG) {
        float s = 0.0f;
        for (int d = 0; d < DDIM; ++d) {
            const float t = Xf[(d >> 1) * (2 * COLS_WG) + tid * 2 + (d & 1)];
            s += t * t;
        }
        x2s[tid] = s;
    } else if (tid >= 64 && tid < 128) {
        const int f = tid - 64;
        float s = 0.0f;
        for (int d = 0; d < DDIM; ++d) {
            const float t = Wf[(d >> 1) * (2 * NFILT) + f * 2 + (d & 1)];
            s += t * t;
        }
        w2s[f] = s;
    }
    __syncthreads();

    // ---- Per-wave 16x16 tile via V_WMMA_F32_16X16X4_F32 ----
    const int wave    = tid >> 5;              // 0..7
    const int lane    = tid & 31;
    const int fBase   = (wave & 3) * 16;       // 4 filter tiles
    const int colBase = (wave >> 2) * 16;      // 2 column tiles

    const int l15  = lane & 15;
    const int half = (lane < 16) ? 0 : 1;      // lanes 0-15: K-pair (4kk,4kk+1); 16-31: (4kk+2,4kk+3)
    const int col  = colBase + l15;            // B-fragment column
    const int fRow = fBase + l15;              // A-fragment row (filter)

    // Loop-invariant fragment bases (float2 units); per-kk delta is a DS offset imm.
    const int aIdx0 = half * NFILT   + fRow;   // +128 float2 (1024 B) per kk
    const int bIdx0 = half * COLS_WG + col;    // + 64 float2 ( 512 B) per kk

    v8f acc = {};
    #pragma unroll
    for (int kk = 0; kk < DPAD / 4; ++kk) {    // 19 WMMA steps over K=76
        const v2f a = WldsP[aIdx0 + kk * (2 * NFILT)];
        const v2f b = XldsP[bIdx0 + kk * (2 * COLS_WG)];
        // 8 args: (neg_a, A, neg_b, B, c_mod, C, reuse_a, reuse_b)
        acc = __builtin_amdgcn_wmma_f32_16x16x4_f32(
                  false, a, false, b, (short)0, acc, false, false);
    }

    // ---- Fuse -0.5*(x2 + w2), deposit tile in LDS for coalesced stores ----
    const float xv   = x2s[col];
    const int   fOff = (lane < 16) ? 0 : 8;    // C/D half-wave M split
    #pragma unroll
    for (int r = 0; r < 8; ++r) {
        const int f = fBase + r + fOff;
        Tile[f * COLS_WG + col] = acc[r] - 0.5f * (xv + w2s[f]);
    }
    __syncthreads();

    // ---- Vectorized output: one 16B granule per (i, f); pBase = 4*wg (16B aligned) ----
    const int pBase = mBase >> 3;              // global spatial base (4 positions per WG)
    for (int g = tid; g < NFILT * N_BATCH; g += 256) {
        const int i = g & 7;
        const int f = g >> 3;
        v4f v;
        v.x = Tile[f * COLS_WG + 0 * N_BATCH + i];
        v.y = Tile[f * COLS_WG + 1 * N_BATCH + i];
        v.z = Tile[f * COLS_WG + 2 * N_BATCH + i];
        v.w = Tile[f * COLS_WG + 3 * N_BATCH + i];
        *(v4f*)(out + i * OUT_ISTR + f * OUT_FSTR + pBase) = v;
    }
}

extern "C" void kernel_launch(void* const* d_in, const int* in_sizes, int n_in,
                              void* d_out, int out_size, void* d_ws, size_t ws_size,
                              hipStream_t stream)
{
    (void)in_sizes; (void)n_in; (void)out_size; (void)d_ws; (void)ws_size;
    const float* x = (const float*)d_in[0];   // (8, 3, 64, 64) fp32
    const float* w = (const float*)d_in[1];   // (64, 3, 5, 5)  fp32
    float* out = (float*)d_out;               // (8, 64, 64, 64) fp32

    const int nBlocks = M_TOTAL / COLS_WG;    // 1024 workgroups, 256 threads (8 waves)
    euclid2d_wmma_f32_kernel<<<nBlocks, 256, 0, stream>>>(x, w, out);
}